// VSModel_82815559401913
// MI455X (gfx1250) — compile-verified
//
#include <hip/hip_runtime.h>
#include <hip/hip_bf16.h>
#include <stdint.h>

// ---------------------------------------------------------------------------
// VSModel log-likelihood:
//   z    = w*v1[bi] + (1-w)*v2[pi]
//   prob = clamp(sigmoid(z), 1e-6, 0.999999)
//   ll   = sum(e1*log(prob) + e2*log(1-prob))
// Memory-bound: 16 B/pair streamed + 2 gathers into 28 KB of tables.
// Strategy: stage v1/v2 into LDS via CDNA5 async-to-LDS (b128), b128
// streaming loads, wave32 shuffle reduction, deterministic 2-kernel reduce.
// ---------------------------------------------------------------------------

#define LL_BLOCKS   1024
#define LL_THREADS  256
#define MAX_NB      5008   // >= 5000, padded
#define MAX_NP      2000

typedef int v4i_async __attribute__((vector_size(16)));
typedef __attribute__((address_space(1))) v4i_async* g_v4i_ptr;
typedef __attribute__((address_space(3))) v4i_async* l_v4i_ptr;

#if __has_builtin(__builtin_amdgcn_global_load_async_to_lds_b128)
#define HAVE_ASYNC_LDS 1
#else
#define HAVE_ASYNC_LDS 0
#endif

__device__ __forceinline__ void wait_async_zero() {
#if __has_builtin(__builtin_amdgcn_s_wait_asynccnt)
    __builtin_amdgcn_s_wait_asynccnt(0);
#else
    asm volatile("s_wait_asynccnt 0" ::: "memory");
#endif
}

__device__ __forceinline__ float pair_term(float b, float p, float w, float omw,
                                           float e1, float e2) {
    float z    = fmaf(w, b, omw * p);
    float prob = 1.0f / (1.0f + __expf(-z));          // v_exp_f32 path
    prob       = fminf(fmaxf(prob, 1e-6f), 0.999999f);
    float lp   = __logf(prob);                        // v_log_f32 path
    float lq   = __logf(1.0f - prob);                 // safe: 1-prob >= 1e-6
    return fmaf(e1, lp, e2 * lq);
}

__global__ void __launch_bounds__(LL_THREADS)
vs_ll_kernel(const float* __restrict__ v1, const float* __restrict__ v2,
             const float* __restrict__ weight,
             const float* __restrict__ e1, const float* __restrict__ e2,
             const int* __restrict__ bi, const int* __restrict__ pi,
             int n, int nb, int np, float* __restrict__ partials) {
    __shared__ __align__(16) float sv1[MAX_NB];
    __shared__ __align__(16) float sv2[MAX_NP];

    const int tid = threadIdx.x;

    // ---- Stage lookup tables into LDS --------------------------------------
#if HAVE_ASYNC_LDS
    {
        const int nb_vec = (nb * 4) & ~15;
        const int np_vec = (np * 4) & ~15;
        for (int off = tid * 16; off < nb_vec; off += LL_THREADS * 16) {
            __builtin_amdgcn_global_load_async_to_lds_b128(
                (g_v4i_ptr)((const char*)v1 + off),
                (l_v4i_ptr)((char*)sv1 + off),
                0, 0);
        }
        for (int off = tid * 16; off < np_vec; off += LL_THREADS * 16) {
            __builtin_amdgcn_global_load_async_to_lds_b128(
                (g_v4i_ptr)((const char*)v2 + off),
                (l_v4i_ptr)((char*)sv2 + off),
                0, 0);
        }
        // tail (< 16 B) via plain path
        for (int i = (nb_vec >> 2) + tid; i < nb; i += LL_THREADS) sv1[i] = v1[i];
        for (int i = (np_vec >> 2) + tid; i < np; i += LL_THREADS) sv2[i] = v2[i];
        wait_async_zero();
    }
#else
    for (int i = tid; i < nb; i += LL_THREADS) sv1[i] = v1[i];
    for (int i = tid; i < np; i += LL_THREADS) sv2[i] = v2[i];
#endif
    __syncthreads();

    const float w   = weight[0];
    const float omw = 1.0f - w;

    // ---- Grid-stride main loop, 4 pairs per iteration (b128 loads) ---------
    const float4* __restrict__ e1v = (const float4*)e1;
    const float4* __restrict__ e2v = (const float4*)e2;
    const int4*   __restrict__ biv = (const int4*)bi;
    const int4*   __restrict__ piv = (const int4*)pi;

    const int nvec   = n >> 2;
    const int stride = LL_BLOCKS * LL_THREADS;

    float acc = 0.0f;
    for (int g = blockIdx.x * LL_THREADS + tid; g < nvec; g += stride) {
        const float4 a  = e1v[g];
        const float4 c  = e2v[g];
        const int4   ib = biv[g];
        const int4   ip = piv[g];
        acc += pair_term(sv1[ib.x], sv2[ip.x], w, omw, a.x, c.x);
        acc += pair_term(sv1[ib.y], sv2[ip.y], w, omw, a.y, c.y);
        acc += pair_term(sv1[ib.z], sv2[ip.z], w, omw, a.z, c.z);
        acc += pair_term(sv1[ib.w], sv2[ip.w], w, omw, a.w, c.w);
    }
    // scalar tail (< 4 elements), deterministic owner
    if (blockIdx.x == 0 && tid == 0) {
        for (int i = nvec << 2; i < n; ++i)
            acc += pair_term(sv1[bi[i]], sv2[pi[i]], w, omw, e1[i], e2[i]);
    }

    // ---- Wave32 + block reduction ------------------------------------------
    for (int off = 16; off > 0; off >>= 1)
        acc += __shfl_xor(acc, off, 32);

    __shared__ float swave[LL_THREADS / 32];
    if ((tid & 31) == 0) swave[tid >> 5] = acc;
    __syncthreads();

    if (tid < 32) {
        float v = (tid < LL_THREADS / 32) ? swave[tid] : 0.0f;
        for (int off = 4; off > 0; off >>= 1)
            v += __shfl_xor(v, off, 32);
        if (tid == 0) partials[blockIdx.x] = v;
    }
}

__global__ void __launch_bounds__(LL_THREADS)
vs_ll_reduce(const float* __restrict__ partials, int n, float* __restrict__ out) {
    const int tid = threadIdx.x;
    float acc = 0.0f;
    for (int i = tid; i < n; i += LL_THREADS) acc += partials[i];

    for (int off = 16; off > 0; off >>= 1)
        acc += __shfl_xor(acc, off, 32);

    __shared__ float swave[LL_THREADS / 32];
    if ((tid & 31) == 0) swave[tid >> 5] = acc;
    __syncthreads();

    if (tid < 32) {
        float v = (tid < LL_THREADS / 32) ? swave[tid] : 0.0f;
        for (int off = 4; off > 0; off >>= 1)
            v += __shfl_xor(v, off, 32);
        if (tid == 0) out[0] = v;
    }
}

extern "C" void kernel_launch(void* const* d_in, const int* in_sizes, int n_in,
                              void* d_out, int out_size, void* d_ws, size_t ws_size,
                              hipStream_t stream) {
    const float* v1 = (const float*)d_in[0];
    const float* v2 = (const float*)d_in[1];
    const float* wt = (const float*)d_in[2];
    const float* e1 = (const float*)d_in[3];
    const float* e2 = (const float*)d_in[4];
    const int*   bi = (const int*)d_in[5];
    const int*   pi = (const int*)d_in[6];

    const int n  = in_sizes[3];
    const int nb = in_sizes[0] < MAX_NB ? in_sizes[0] : MAX_NB;
    const int np = in_sizes[1] < MAX_NP ? in_sizes[1] : MAX_NP;

    float* partials = (float*)d_ws;  // LL_BLOCKS floats = 4 KB

    vs_ll_kernel<<<LL_BLOCKS, LL_THREADS, 0, stream>>>(
        v1, v2, wt, e1, e2, bi, pi, n, nb, np, partials);
    vs_ll_reduce<<<1, LL_THREADS, 0, stream>>>(partials, LL_BLOCKS, (float*)d_out);
}